// SelfAttention_90340342104707
// MI455X (gfx1250) — compile-verified
//
#include <hip/hip_runtime.h>
#include <hip/hip_bf16.h>

#define FDIM 4096
#define BSZ  8
#define CDIM 512

typedef __attribute__((ext_vector_type(2))) float v2f;
typedef __attribute__((ext_vector_type(8))) float v8f;

// ---------------------------------------------------------------------------
// Kernel 0: reduce A = wq.wk and Bk = bq.wk, pre-scaled by log2(e).
// (softmax over j is shift-invariant, so only these two scalars survive:
//  attention[b,i,:] = softmax_j( (A*x_i + Bk) * x_j ) )
// ---------------------------------------------------------------------------
__global__ __launch_bounds__(256) void precompute_scalars(
    const float* __restrict__ wq, const float* __restrict__ bq,
    const float* __restrict__ wk, float* __restrict__ scal) {
  __shared__ float red0[256];
  __shared__ float red1[256];
  const int t = threadIdx.x;
  float a = 0.f, bk = 0.f;
  for (int c = t; c < CDIM; c += 256) {
    const float wkc = wk[c];
    a  += wq[c] * wkc;
    bk += bq[c] * wkc;
  }
  red0[t] = a; red1[t] = bk;
  __syncthreads();
  for (int off = 128; off > 0; off >>= 1) {
    if (t < off) { red0[t] += red0[t + off]; red1[t] += red1[t + off]; }
    __syncthreads();
  }
  if (t == 0) {
    const float L2E = 1.44269504088896340736f;
    scal[0] = red0[0] * L2E;   // A' = (wq.wk)*log2e
    scal[1] = red1[0] * L2E;   // Bk' = (bq.wk)*log2e
  }
}

// ---------------------------------------------------------------------------
// half-wave swap (lane L <-> lane L^16) via LDS swizzle unit (no LDS memory)
// ---------------------------------------------------------------------------
__device__ __forceinline__ float swapx16(float v) {
  int i = __float_as_int(v);
  i = __builtin_amdgcn_ds_swizzle(i, 0x401F);  // group-of-32, xor=0x10, and=0x1f
  return __int_as_float(i);
}

// low 32 bits of a generic pointer to __shared__ == DS-space byte address
__device__ __forceinline__ uint32_t lds_addr_of(const void* p) {
  return (uint32_t)(uintptr_t)p;
}

// ---------------------------------------------------------------------------
// Kernel 1: flash-style softmax over j with rank-1 energy tiles via WMMA.
// Grid: 2048 blocks = 8 batches * 256 i-tiles; 128 threads = 4 waves.
//  - async-copy the batch's x row (16KB) into LDS (GLOBAL_LOAD_ASYNC_TO_LDS)
//  - build v row (wv*x+bv) in LDS
//  - each wave: E^T tile [j x i] = x_j * s'_i via v_wmma_f32_16x16x4_f32
//    (K=0 only), exp2 the 8 accumulator VGPRs, fold into per-lane num/den.
// ---------------------------------------------------------------------------
__global__ __launch_bounds__(128) void attn_main(
    const float* __restrict__ x, const float* __restrict__ scal,
    const float* __restrict__ wvp, const float* __restrict__ bvp,
    const float* __restrict__ gamma, float* __restrict__ out) {
  __shared__ __align__(16) float lds_x[FDIM];
  __shared__ __align__(16) float lds_v[FDIM];
  __shared__ float sh_num[16];
  __shared__ float sh_den[16];

  const int tid    = threadIdx.x;
  const int b      = blockIdx.x >> 8;          // 256 i-tiles per batch
  const int i_base = (blockIdx.x & 255) << 4;
  const float* __restrict__ xb = x + b * FDIM;

  // ---- async copy x row into LDS: 1024 x 16B chunks over 128 threads ----
#pragma unroll
  for (int k = 0; k < 8; ++k) {
    const int c = tid + (k << 7);                       // chunk 0..1023
    const uint32_t la = lds_addr_of(&lds_x[c << 2]);
    const float* ga = xb + (c << 2);
    asm volatile("global_load_async_to_lds_b128 %0, %1, off"
                 :: "v"(la), "v"(ga) : "memory");
  }
  asm volatile("s_wait_asynccnt 0x0" ::: "memory");
  if (tid < 16) { sh_num[tid] = 0.f; sh_den[tid] = 0.f; }
  __syncthreads();

  // ---- build v row in LDS: v_j = wv*x_j + bv ----
  const float wv0 = wvp[0];
  const float bv0 = bvp[0];
#pragma unroll
  for (int k = 0; k < 8; ++k) {
    const int c = tid + (k << 7);
    const float4 xv = *reinterpret_cast<const float4*>(&lds_x[c << 2]);
    float4 vv;
    vv.x = wv0 * xv.x + bv0;
    vv.y = wv0 * xv.y + bv0;
    vv.z = wv0 * xv.z + bv0;
    vv.w = wv0 * xv.w + bv0;
    *reinterpret_cast<float4*>(&lds_v[c << 2]) = vv;
  }
  __syncthreads();

  const int lane = tid & 31;
  const int wave = tid >> 5;
  const int lo = lane & 15;
  const int hi = lane >> 4;       // which half-wave

  // B operand: K=0 row = s'_i = A'*x_i + Bk' (lanes 0-15), everything else 0.
  const float sval = scal[0] * lds_x[i_base + lo] + scal[1];
  v2f Bm; Bm.x = hi ? 0.f : sval; Bm.y = 0.f;

  float num = 0.f, den = 0.f;

  for (int jt = wave; jt < FDIM / 16; jt += 4) {
    const int j_base = jt << 4;
    // A operand: K=0 column = x_j (lanes 0-15 hold rows M=j_local).
    const float xval = lds_x[j_base + lo];
    v2f Am; Am.x = hi ? 0.f : xval; Am.y = 0.f;

    v8f Cm = {};
    // D[m][n] = x_{j_base+m} * s'_{i_base+n}  (log2-scaled energy, transposed)
    v8f D = __builtin_amdgcn_wmma_f32_16x16x4_f32(
        false, Am, false, Bm, (short)0, Cm, false, false);

    // v values for the 8 j-rows this lane holds (C layout: M = r + 8*hi).
    const float4* vp =
        reinterpret_cast<const float4*>(&lds_v[j_base + (hi << 3)]);
    const float4 v0 = vp[0];
    const float4 v1 = vp[1];

    float p;
    p = __builtin_amdgcn_exp2f(D[0]); num += p * v0.x; den += p;
    p = __builtin_amdgcn_exp2f(D[1]); num += p * v0.y; den += p;
    p = __builtin_amdgcn_exp2f(D[2]); num += p * v0.z; den += p;
    p = __builtin_amdgcn_exp2f(D[3]); num += p * v0.w; den += p;
    p = __builtin_amdgcn_exp2f(D[4]); num += p * v1.x; den += p;
    p = __builtin_amdgcn_exp2f(D[5]); num += p * v1.y; den += p;
    p = __builtin_amdgcn_exp2f(D[6]); num += p * v1.z; den += p;
    p = __builtin_amdgcn_exp2f(D[7]); num += p * v1.w; den += p;
  }

  // combine the two half-waves: lane L holds j in {0..7}+16k, lane L+16 {8..15}+16k
  num += swapx16(num);
  den += swapx16(den);

  if (lane < 16) {
    atomicAdd(&sh_num[lo], num);   // ds_add_f32
    atomicAdd(&sh_den[lo], den);
  }
  __syncthreads();

  if (tid < 16) {
    const int i = i_base + tid;
    out[b * FDIM + i] = gamma[0] * (sh_num[tid] / sh_den[tid]) + lds_x[i];
  }
}

// ---------------------------------------------------------------------------
extern "C" void kernel_launch(void* const* d_in, const int* in_sizes, int n_in,
                              void* d_out, int out_size, void* d_ws, size_t ws_size,
                              hipStream_t stream) {
  const float* x     = (const float*)d_in[0];
  const float* wq    = (const float*)d_in[1];
  const float* bq    = (const float*)d_in[2];
  const float* wk    = (const float*)d_in[3];
  const float* bk    = (const float*)d_in[4];  // unused: cancels in softmax
  const float* wv    = (const float*)d_in[5];
  const float* bv    = (const float*)d_in[6];
  const float* gamma = (const float*)d_in[7];
  (void)bk; (void)in_sizes; (void)n_in; (void)out_size; (void)ws_size;

  float* out  = (float*)d_out;
  float* scal = (float*)d_ws;   // 2 floats

  precompute_scalars<<<1, 256, 0, stream>>>(wq, bq, wk, scal);
  attn_main<<<BSZ * (FDIM / 16), 128, 0, stream>>>(x, scal, wv, bv, gamma, out);
}